// CompetitiveLayer_85736137163123
// MI455X (gfx1250) — compile-verified
//
#include <hip/hip_runtime.h>

typedef float v2f __attribute__((ext_vector_type(2)));
typedef float v4f __attribute__((ext_vector_type(4)));
typedef float v8f __attribute__((ext_vector_type(8)));

#ifndef N_ITERS
#define N_ITERS 20
#endif

__device__ __forceinline__ v2f mk2(float x, float y) { v2f v; v[0] = x; v[1] = y; return v; }

// D = A(16x4 f32) * B(4x16 f32) + C(16x16 f32), wave32 WMMA
__device__ __forceinline__ v8f wmma4(v2f a, v2f b, v8f c) {
  return __builtin_amdgcn_wmma_f32_16x16x4_f32(false, a, false, b, (short)0, c, false, false);
}

__global__ __launch_bounds__(256)
void comp_eq_wmma(const float* __restrict__ AT, const float* __restrict__ kmat,
                  const float* __restrict__ btv, float* __restrict__ out, int n_pairs)
{
  const int lane = threadIdx.x & 31;
  const int r    = lane & 15;      // row within 16-row tile / A-matrix M
  const int h    = lane >> 4;      // lane half (selects feature pair)
  // A rows 8..15 hold K rows permuted by sigma(m)=m^2 so the WMMA D upper
  // lane-half lands features {2,3,6,7} exactly where the next B-operand
  // wants them (no cross-lane shuffles in the hot loop).
  const int am = (r < 8) ? r : ((r - 8) ^ 2);

  // Per-lane constant A operands: K and K^T in WMMA A-layout (feature f = 4s+2h+v)
  const v2f aK0  = mk2(__expf(kmat[am * 8 + 2 * h    ]), __expf(kmat[am * 8 + 2 * h + 1]));
  const v2f aK1  = mk2(__expf(kmat[am * 8 + 2 * h + 4]), __expf(kmat[am * 8 + 2 * h + 5]));
  const v2f aKT0 = mk2(__expf(kmat[(2 * h    ) * 8 + am]), __expf(kmat[(2 * h + 1) * 8 + am]));
  const v2f aKT1 = mk2(__expf(kmat[(2 * h + 4) * 8 + am]), __expf(kmat[(2 * h + 5) * 8 + am]));

  // BT = exp(bt) in B-operand layout
  const float BT00 = __expf(btv[2 * h]);
  const float BT01 = __expf(btv[2 * h + 1]);
  const float BT10 = __expf(btv[2 * h + 4]);
  const float BT11 = __expf(btv[2 * h + 5]);

  v8f ones;
#pragma unroll
  for (int i = 0; i < 8; ++i) ones[i] = 1.0f;   // folds the "+1" (compiler emits inline 1.0)

  const int wid = (int)((blockIdx.x * blockDim.x + threadIdx.x) >> 5);
  const int nw  = (int)((gridDim.x * blockDim.x) >> 5);

  for (int tp = wid; tp < n_pairs; tp += nw) {
    // Two adjacent 16-row tiles per wave, interleaved for in-wave ILP.
    const size_t rowA = (size_t)(2 * tp    ) * 16 + r;
    const size_t rowB = (size_t)(2 * tp + 1) * 16 + r;
    const float* atpA = AT + rowA * 8;
    const float* atpB = AT + rowB * 8;
    const v2f a0A = *(const v2f*)(atpA + 2 * h);       // AT in B-layout, step 0
    const v2f a1A = *(const v2f*)(atpA + 2 * h + 4);   // step 1
    const v2f a0B = *(const v2f*)(atpB + 2 * h);
    const v2f a1B = *(const v2f*)(atpB + 2 * h + 4);

    float BFA00 = BT00, BFA01 = BT01, BFA10 = BT10, BFA11 = BT11;  // BF init = BTb
    float BFB00 = BT00, BFB01 = BT01, BFB10 = BT10, BFB11 = BT11;
    float AFA00 = 0.f, AFA01 = 0.f, AFA10 = 0.f, AFA11 = 0.f;
    float AFB00 = 0.f, AFB01 = 0.f, AFB10 = 0.f, AFB11 = 0.f;

    for (int it = 0; it < N_ITERS; ++it) {
      // AFden^T = K * BF^T + 1 (two chained k-steps), tiles A and B interleaved
      v8f dA = wmma4(aK0, mk2(BFA00, BFA01), ones);
      v8f dB = wmma4(aK0, mk2(BFB00, BFB01), ones);
      dA = wmma4(aK1, mk2(BFA10, BFA11), dA);
      dB = wmma4(aK1, mk2(BFB10, BFB11), dB);
      AFA00 = a0A[0] * __builtin_amdgcn_rcpf(dA[0]);
      AFB00 = a0B[0] * __builtin_amdgcn_rcpf(dB[0]);
      AFA01 = a0A[1] * __builtin_amdgcn_rcpf(dA[1]);
      AFB01 = a0B[1] * __builtin_amdgcn_rcpf(dB[1]);
      AFA10 = a1A[0] * __builtin_amdgcn_rcpf(dA[4]);
      AFB10 = a1B[0] * __builtin_amdgcn_rcpf(dB[4]);
      AFA11 = a1A[1] * __builtin_amdgcn_rcpf(dA[5]);
      AFB11 = a1B[1] * __builtin_amdgcn_rcpf(dB[5]);
      // BFden^T = K^T * AF^T + 1
      v8f eA = wmma4(aKT0, mk2(AFA00, AFA01), ones);
      v8f eB = wmma4(aKT0, mk2(AFB00, AFB01), ones);
      eA = wmma4(aKT1, mk2(AFA10, AFA11), eA);
      eB = wmma4(aKT1, mk2(AFB10, AFB11), eB);
      BFA00 = BT00 * __builtin_amdgcn_rcpf(eA[0]);
      BFB00 = BT00 * __builtin_amdgcn_rcpf(eB[0]);
      BFA01 = BT01 * __builtin_amdgcn_rcpf(eA[1]);
      BFB01 = BT01 * __builtin_amdgcn_rcpf(eB[1]);
      BFA10 = BT10 * __builtin_amdgcn_rcpf(eA[4]);
      BFB10 = BT10 * __builtin_amdgcn_rcpf(eB[4]);
      BFA11 = BT11 * __builtin_amdgcn_rcpf(eA[5]);
      BFB11 = BT11 * __builtin_amdgcn_rcpf(eB[5]);
    }

    // Final correction: AF_ from final BF; BF_ from pre-correction AF
    v8f dA = wmma4(aK0, mk2(BFA00, BFA01), ones);
    v8f dB = wmma4(aK0, mk2(BFB00, BFB01), ones);
    dA = wmma4(aK1, mk2(BFA10, BFA11), dA);
    dB = wmma4(aK1, mk2(BFB10, BFB11), dB);
    const float AfA00 = a0A[0] * __builtin_amdgcn_rcpf(dA[0]);
    const float AfB00 = a0B[0] * __builtin_amdgcn_rcpf(dB[0]);
    const float AfA01 = a0A[1] * __builtin_amdgcn_rcpf(dA[1]);
    const float AfB01 = a0B[1] * __builtin_amdgcn_rcpf(dB[1]);
    const float AfA10 = a1A[0] * __builtin_amdgcn_rcpf(dA[4]);
    const float AfB10 = a1B[0] * __builtin_amdgcn_rcpf(dB[4]);
    const float AfA11 = a1A[1] * __builtin_amdgcn_rcpf(dA[5]);
    const float AfB11 = a1B[1] * __builtin_amdgcn_rcpf(dB[5]);
    v8f eA = wmma4(aKT0, mk2(AFA00, AFA01), ones);
    v8f eB = wmma4(aKT0, mk2(AFB00, AFB01), ones);
    eA = wmma4(aKT1, mk2(AFA10, AFA11), eA);
    eB = wmma4(aKT1, mk2(AFB10, AFB11), eB);
    const float BfA00 = BT00 * __builtin_amdgcn_rcpf(eA[0]);
    const float BfB00 = BT00 * __builtin_amdgcn_rcpf(eB[0]);
    const float BfA01 = BT01 * __builtin_amdgcn_rcpf(eA[1]);
    const float BfB01 = BT01 * __builtin_amdgcn_rcpf(eB[1]);
    const float BfA10 = BT10 * __builtin_amdgcn_rcpf(eA[4]);
    const float BfB10 = BT10 * __builtin_amdgcn_rcpf(eB[4]);
    const float BfA11 = BT11 * __builtin_amdgcn_rcpf(eA[5]);
    const float BfB11 = BT11 * __builtin_amdgcn_rcpf(eB[5]);

    // K rows 4h..4h+3 for the outer product — computed here (accumulator temps
    // are dead now) to keep hot-loop register pressure low.
    float kout[32];
#pragma unroll
    for (int i = 0; i < 8; ++i) {
      v4f kk = *(const v4f*)(kmat + 32 * h + 4 * i);
#pragma unroll
      for (int j = 0; j < 4; ++j) kout[4 * i + j] = __expf(kk[j]);
    }

#pragma unroll
    for (int t = 0; t < 2; ++t) {
      const float Af00 = t ? AfB00 : AfA00, Af01 = t ? AfB01 : AfA01;
      const float Af10 = t ? AfB10 : AfA10, Af11 = t ? AfB11 : AfA11;
      const float Bf00 = t ? BfB00 : BfA00, Bf01 = t ? BfB01 : BfA01;
      const float Bf10 = t ? BfB10 : BfA10, Bf11 = t ? BfB11 : BfA11;
      const size_t row = t ? rowB : rowA;

      // One-time cross-half exchange to assemble outer-product operands
      const float SAf00 = __shfl_xor(Af00, 16, 32);
      const float SAf01 = __shfl_xor(Af01, 16, 32);
      const float SAf10 = __shfl_xor(Af10, 16, 32);
      const float SAf11 = __shfl_xor(Af11, 16, 32);
      const float SBf00 = __shfl_xor(Bf00, 16, 32);
      const float SBf01 = __shfl_xor(Bf01, 16, 32);
      const float SBf10 = __shfl_xor(Bf10, 16, 32);
      const float SBf11 = __shfl_xor(Bf11, 16, 32);

      float afo[4], bfo[8];
      afo[0] = h ? SAf10 : Af00;   // a = 4h+0
      afo[1] = h ? SAf11 : Af01;   // a = 4h+1
      afo[2] = h ? Af10  : SAf00;  // a = 4h+2
      afo[3] = h ? Af11  : SAf01;  // a = 4h+3
      bfo[0] = h ? SBf00 : Bf00;
      bfo[1] = h ? SBf01 : Bf01;
      bfo[2] = h ? Bf00  : SBf00;
      bfo[3] = h ? Bf01  : SBf01;
      bfo[4] = h ? SBf10 : Bf10;
      bfo[5] = h ? SBf11 : Bf11;
      bfo[6] = h ? Bf10  : SBf10;
      bfo[7] = h ? Bf11  : SBf11;

      // C[row, a, b] = AF_[row,a] * BF_[row,b] * K[a,b]; lane writes a in [4h, 4h+4)
      float* outp = out + row * 64 + (size_t)h * 32;
#pragma unroll
      for (int a = 0; a < 4; ++a) {
        v4f c0, c1;
#pragma unroll
        for (int b = 0; b < 4; ++b) {
          c0[b] = afo[a] * bfo[b]     * kout[a * 8 + b];
          c1[b] = afo[a] * bfo[b + 4] * kout[a * 8 + b + 4];
        }
        *(v4f*)(outp + a * 8)     = c0;
        *(v4f*)(outp + a * 8 + 4) = c1;
      }
    }
  }
}

extern "C" void kernel_launch(void* const* d_in, const int* in_sizes, int n_in,
                              void* d_out, int out_size, void* d_ws, size_t ws_size,
                              hipStream_t stream) {
  const float* AT = (const float*)d_in[0];   // [B, 8]
  const float* k  = (const float*)d_in[1];   // [8, 8]
  const float* bt = (const float*)d_in[2];   // [1, 8]
  float* out = (float*)d_out;                // [B, 8, 8]

  const int rows    = in_sizes[0] / 8;
  const int n_pairs = rows / 32;             // 16384 tile-pairs for B=524288

  dim3 grid(2048), block(256);               // 16384 waves -> one tile-pair each
  hipLaunchKernelGGL(comp_eq_wmma, grid, block, 0, stream, AT, k, bt, out, n_pairs);
}